// LorentzInteractionNetwork_69861938037253
// MI455X (gfx1250) — compile-verified
//
#include <hip/hip_runtime.h>
#include <hip/hip_bf16.h>

#define HID 14
#define OUTD 2

typedef float v2f __attribute__((ext_vector_type(2)));
typedef float v8f __attribute__((ext_vector_type(8)));

__device__ __forceinline__ float minkip(float4 a, float4 b) {
  return -a.x * b.x + a.y * b.y + a.z * b.z + a.w * b.w;
}
__device__ __forceinline__ float psi_fn(float v) {
  return copysignf(__logf(fabsf(v) + 1.0f), v);
}

__device__ __forceinline__ v8f wmma4(v2f a, v2f b, v8f c) {
  return __builtin_amdgcn_wmma_f32_16x16x4_f32(false, a, false, b, (short)0, c,
                                               false, false);
}

// ---------------------------------------------------------------------------
// K=16 transposed GEMM: D' = A(16x16,W^T) @ Z(16x16) where Z rows = previous
// activation rows held in WMMA C-layout:
//   lane n (n<16):  d[v] = row v     of column n
//   lane n (n>=16): d[v] = row v + 8 of column n-16
// Upper half-wave needs lower's d[2],d[3],d[6],d[7]; lower needs upper's
// d[0],d[1],d[4],d[5] -- one shfl_xor(16) per pair serves BOTH directions.
// ---------------------------------------------------------------------------
__device__ __forceinline__ v8f gemm16(const v2f A[4], const float d[8],
                                      const float bias[8], bool low) {
  float t0 = __shfl_xor(low ? d[2] : d[0], 16);
  float t1 = __shfl_xor(low ? d[3] : d[1], 16);
  float t2 = __shfl_xor(low ? d[6] : d[4], 16);
  float t3 = __shfl_xor(low ? d[7] : d[5], 16);
  v8f acc;
#pragma unroll
  for (int v = 0; v < 8; ++v) acc[v] = bias[v];
  v2f b;
  b.x = low ? d[0] : t0;  b.y = low ? d[1] : t1;  acc = wmma4(A[0], b, acc);
  b.x = low ? d[4] : t2;  b.y = low ? d[5] : t3;  acc = wmma4(A[1], b, acc);
  b.x = low ? t0 : d[2];  b.y = low ? t1 : d[3];  acc = wmma4(A[2], b, acc);
  b.x = low ? t2 : d[6];  b.y = low ? t3 : d[7];  acc = wmma4(A[3], b, acc);
  return acc;
}

// Same, but the logical z-vector is [z0, d0..d13, 0]: row 0 is an extra
// per-column scalar held by the lower half-wave.
__device__ __forceinline__ v8f gemm16_off(const v2f A[4], const float d[8],
                                          float z0, const float bias[8],
                                          bool low) {
  float t0 = __shfl_xor(low ? d[1] : d[0], 16);  // up<-low d1 ; low<-up d0 (=z9)
  float t1 = __shfl_xor(low ? d[2] : d[3], 16);  // up<-low d2 ; low<-up d3 (=z12)
  float t2 = __shfl_xor(low ? d[5] : d[4], 16);  // up<-low d5 ; low<-up d4 (=z13)
  float t3 = __shfl_xor(d[6], 16);               // up<-low d6
  v8f acc;
#pragma unroll
  for (int v = 0; v < 8; ++v) acc[v] = bias[v];
  v2f b;
  b.x = low ? z0   : t0;   b.y = low ? d[0] : t1;  acc = wmma4(A[0], b, acc);
  b.x = low ? d[3] : t2;   b.y = low ? d[4] : t3;  acc = wmma4(A[1], b, acc);
  b.x = low ? d[7] : d[1]; b.y = low ? t0 : d[2];  acc = wmma4(A[2], b, acc);
  b.x = low ? t1 : d[5];   b.y = low ? t2 : d[6];  acc = wmma4(A[3], b, acc);
  return acc;
}

// ---------------------------------------------------------------------------
// Edge kernel: 16 edges per wave-tile; edge MLP (4->14->14) + message MLP
// (15->14->14) via transposed 16x16x4 f32 WMMAs, atomic scatter to agg[col].
// ---------------------------------------------------------------------------
__global__ void lin_edge_kernel(
    const float* __restrict__ x, const int* __restrict__ row,
    const int* __restrict__ col,
    const float* __restrict__ We1, const float* __restrict__ be1,
    const float* __restrict__ We2, const float* __restrict__ be2,
    const float* __restrict__ Wn11, const float* __restrict__ bn11,
    const float* __restrict__ Wn12, const float* __restrict__ bn12,
    float* __restrict__ aggS, float* __restrict__ aggC, int E, int nTiles) {
  const int lane = threadIdx.x & 31;
  const bool low = lane < 16;
  const int h = lane & 15;  // hidden row for A operands / edge slot in tile
  const int wid = (blockIdx.x * blockDim.x + threadIdx.x) >> 5;
  const int nWaves = (gridDim.x * blockDim.x) >> 5;
  const int k0 = low ? 0 : 2;

  // A operands (transposed weights), loaded once per wave.
  v2f a1;
  a1.x = (h < HID) ? We1[(k0 + 0) * HID + h] : 0.f;
  a1.y = (h < HID) ? We1[(k0 + 1) * HID + h] : 0.f;
  v2f aE2[4], aN1[4], aN2[4];
#pragma unroll
  for (int k = 0; k < 4; ++k) {
    int kk = 4 * k + k0;
    aE2[k].x = (h < HID && kk + 0 < HID)     ? We2 [(kk + 0) * HID + h] : 0.f;
    aE2[k].y = (h < HID && kk + 1 < HID)     ? We2 [(kk + 1) * HID + h] : 0.f;
    aN1[k].x = (h < HID && kk + 0 < HID + 1) ? Wn11[(kk + 0) * HID + h] : 0.f;
    aN1[k].y = (h < HID && kk + 1 < HID + 1) ? Wn11[(kk + 1) * HID + h] : 0.f;
    aN2[k].x = (h < HID && kk + 0 < HID)     ? Wn12[(kk + 0) * HID + h] : 0.f;
    aN2[k].y = (h < HID && kk + 1 < HID)     ? Wn12[(kk + 1) * HID + h] : 0.f;
  }
  float bE1[8], bE2[8], bN1[8], bN2[8];
#pragma unroll
  for (int v = 0; v < 8; ++v) {
    int m = v + (low ? 0 : 8);
    bE1[v] = (m < HID) ? be1[m] : 0.f;
    bE2[v] = (m < HID) ? be2[m] : 0.f;
    bN1[v] = (m < HID) ? bn11[m] : 0.f;
    bN2[v] = (m < HID) ? bn12[m] : 0.f;
  }
  const float4* x4 = (const float4*)x;

  for (int tile = wid; tile < nTiles; tile += nWaves) {
    int e = tile * 16 + h;
    bool valid = e < E;

    // Wave-uniform prefetch of next tile's indices (global_prefetch_b8).
    if (tile + nWaves < nTiles) {
      int en = (tile + nWaves) * 16 + h;
      __builtin_prefetch(row + en, 0, 0);
      __builtin_prefetch(col + en, 0, 0);
    }

    int ce = 0;
    float f0 = 0.f, f1 = 0.f, ip_ss = 0.f;
    if (valid) {
      int r = row[e];
      ce = col[e];
      float4 s = x4[r], dd = x4[ce];
      if (low) {
        ip_ss = minkip(s, s);
        f0 = ip_ss;
        f1 = minkip(s, dd);
      } else {
        f0 = psi_fn(minkip(dd, dd));
        float4 sd = make_float4(s.x - dd.x, s.y - dd.y, s.z - dd.z, s.w - dd.w);
        f1 = psi_fn(minkip(sd, sd));
      }
    }

    // GEMM1: h1^T = We1^T @ efeat^T   (one 16x16x4 WMMA)
    v2f b; b.x = f0; b.y = f1;
    v8f acc;
#pragma unroll
    for (int v = 0; v < 8; ++v) acc[v] = bE1[v];
    acc = wmma4(a1, b, acc);
    float d1[8];
#pragma unroll
    for (int v = 0; v < 8; ++v) d1[v] = fmaxf(acc[v], 0.f);

    // GEMM2: edge_attr^T = We2^T @ relu(h1)^T
    acc = gemm16(aE2, d1, bE2, low);
    float d2[8];
#pragma unroll
    for (int v = 0; v < 8; ++v) d2[v] = acc[v];

    // GEMM3: node MLP1 layer 1 on z = [ip_ss, edge_attr]
    acc = gemm16_off(aN1, d2, ip_ss, bN1, low);
    float d3[8];
#pragma unroll
    for (int v = 0; v < 8; ++v) d3[v] = fmaxf(acc[v], 0.f);

    // GEMM4: m^T = Wn12^T @ relu(...)
    acc = gemm16(aN2, d3, bN2, low);

    if (valid) {
      float* dstp = aggS + (size_t)ce * HID;
      if (low) {
#pragma unroll
        for (int v = 0; v < 8; ++v) atomicAdd(dstp + v, acc[v]);
        atomicAdd(aggC + ce, 1.0f);
      } else {
#pragma unroll
        for (int v = 0; v < 6; ++v) atomicAdd(dstp + 8 + v, acc[v]);
      }
    }
  }
}

// ---------------------------------------------------------------------------
// Node kernel: x_out = MLP([ip(x,x), agg]) with the same WMMA scheme, scatter
// into per-graph sums.
// ---------------------------------------------------------------------------
__global__ void lin_node_kernel(
    const float* __restrict__ x, const int* __restrict__ batch,
    const float* __restrict__ aggS, const float* __restrict__ aggC,
    const float* __restrict__ Wn21, const float* __restrict__ bn21,
    const float* __restrict__ Wn22, const float* __restrict__ bn22,
    float* __restrict__ gS, float* __restrict__ gC, int Nn, int nTiles) {
  const int lane = threadIdx.x & 31;
  const bool low = lane < 16;
  const int h = lane & 15;
  const int wid = (blockIdx.x * blockDim.x + threadIdx.x) >> 5;
  const int nWaves = (gridDim.x * blockDim.x) >> 5;
  const int k0 = low ? 0 : 2;

  v2f aL1[4], aL2[4];
#pragma unroll
  for (int k = 0; k < 4; ++k) {
    int kk = 4 * k + k0;
    aL1[k].x = (h < HID && kk + 0 < HID + 1) ? Wn21[(kk + 0) * HID + h] : 0.f;
    aL1[k].y = (h < HID && kk + 1 < HID + 1) ? Wn21[(kk + 1) * HID + h] : 0.f;
    aL2[k].x = (h < HID && kk + 0 < HID)     ? Wn22[(kk + 0) * HID + h] : 0.f;
    aL2[k].y = (h < HID && kk + 1 < HID)     ? Wn22[(kk + 1) * HID + h] : 0.f;
  }
  float bL1[8], bL2[8];
#pragma unroll
  for (int v = 0; v < 8; ++v) {
    int m = v + (low ? 0 : 8);
    bL1[v] = (m < HID) ? bn21[m] : 0.f;
    bL2[v] = (m < HID) ? bn22[m] : 0.f;
  }
  const float4* x4 = (const float4*)x;

  for (int tile = wid; tile < nTiles; tile += nWaves) {
    int nidx = tile * 16 + h;
    bool valid = nidx < Nn;

    if (tile + nWaves < nTiles) {
      int nn = (tile + nWaves) * 16 + h;
      __builtin_prefetch(aggS + (size_t)nn * HID, 0, 0);
      __builtin_prefetch(aggC + nn, 0, 0);
    }

    float ipxx = 0.f, inv = 0.f;
    int bg = 0;
    const float* srow = aggS + (size_t)(valid ? nidx : 0) * HID;
    if (valid) {
      float4 xv = x4[nidx];
      ipxx = minkip(xv, xv);
      inv = 1.0f / fmaxf(aggC[nidx], 1.0f);
      bg = batch[nidx];
    }
    v8f acc;
#pragma unroll
    for (int v = 0; v < 8; ++v) acc[v] = bL1[v];
#pragma unroll
    for (int k = 0; k < 4; ++k) {
      int j0 = 4 * k + (low ? 0 : 2);
      float b0 = 0.f, b1v = 0.f;
      if (valid) {
        b0 = (j0 == 0) ? ipxx : ((j0 <= HID) ? srow[j0 - 1] * inv : 0.f);
        int j1 = j0 + 1;
        b1v = (j1 <= HID) ? srow[j1 - 1] * inv : 0.f;
      }
      v2f bb; bb.x = b0; bb.y = b1v;
      acc = wmma4(aL1[k], bb, acc);
    }
    float d1[8];
#pragma unroll
    for (int v = 0; v < 8; ++v) d1[v] = fmaxf(acc[v], 0.f);

    acc = gemm16(aL2, d1, bL2, low);

    if (valid) {
      float* dstp = gS + (size_t)bg * HID;
      if (low) {
#pragma unroll
        for (int v = 0; v < 8; ++v) atomicAdd(dstp + v, acc[v]);
        atomicAdd(gC + bg, 1.0f);
      } else {
#pragma unroll
        for (int v = 0; v < 6; ++v) atomicAdd(dstp + 8 + v, acc[v]);
      }
    }
  }
}

// ---------------------------------------------------------------------------
// Global kernel: per-graph mean + MLP 14->14->2 (tiny; scalar VALU is fine).
// ---------------------------------------------------------------------------
__global__ void lin_global_kernel(const float* __restrict__ gS,
                                  const float* __restrict__ gC,
                                  const float* __restrict__ Wg1,
                                  const float* __restrict__ bg1,
                                  const float* __restrict__ Wg2,
                                  const float* __restrict__ bg2,
                                  float* __restrict__ out, int G) {
  int g = blockIdx.x * blockDim.x + threadIdx.x;
  if (g >= G) return;
  float inv = 1.0f / fmaxf(gC[g], 1.0f);
  float zv[HID], h1[HID];
#pragma unroll
  for (int j = 0; j < HID; ++j) zv[j] = gS[(size_t)g * HID + j] * inv;
#pragma unroll
  for (int j2 = 0; j2 < HID; ++j2) {
    float s = bg1[j2];
#pragma unroll
    for (int j = 0; j < HID; ++j) s += zv[j] * Wg1[j * HID + j2];
    h1[j2] = fmaxf(s, 0.f);
  }
#pragma unroll
  for (int o = 0; o < OUTD; ++o) {
    float s = bg2[o];
#pragma unroll
    for (int j = 0; j < HID; ++j) s += h1[j] * Wg2[j * OUTD + o];
    out[(size_t)g * OUTD + o] = s;
  }
}

static inline int imin(int a, int b) { return a < b ? a : b; }

extern "C" void kernel_launch(void* const* d_in, const int* in_sizes, int n_in,
                              void* d_out, int out_size, void* d_ws,
                              size_t ws_size, hipStream_t stream) {
  const float* x     = (const float*)d_in[0];
  const int*   ei    = (const int*)d_in[1];
  const int*   batch = (const int*)d_in[2];
  const float* We1 = (const float*)d_in[3],  *be1 = (const float*)d_in[4];
  const float* We2 = (const float*)d_in[5],  *be2 = (const float*)d_in[6];
  const float* Wn11 = (const float*)d_in[7], *bn11 = (const float*)d_in[8];
  const float* Wn12 = (const float*)d_in[9], *bn12 = (const float*)d_in[10];
  const float* Wn21 = (const float*)d_in[11], *bn21 = (const float*)d_in[12];
  const float* Wn22 = (const float*)d_in[13], *bn22 = (const float*)d_in[14];
  const float* Wg1 = (const float*)d_in[15], *bg1 = (const float*)d_in[16];
  const float* Wg2 = (const float*)d_in[17], *bg2 = (const float*)d_in[18];

  const int Nn = in_sizes[0] / 4;
  const int E  = in_sizes[1] / 2;
  const int G  = out_size / OUTD;
  const int* row = ei;
  const int* col = ei + E;

  // Workspace: agg sums [N,H], agg counts [N], graph sums [G,H], counts [G]
  float* aggS = (float*)d_ws;
  float* aggC = aggS + (size_t)Nn * HID;
  float* gS   = aggC + Nn;
  float* gC   = gS + (size_t)G * HID;
  size_t wsNeed =
      ((size_t)Nn * (HID + 1) + (size_t)G * (HID + 1)) * sizeof(float);
  hipMemsetAsync(d_ws, 0, wsNeed, stream);

  dim3 blk(256);  // 8 wave32 waves per block
  const int wpb = 256 / 32;

  int nTilesE = (E + 15) / 16;
  int gridE = imin((nTilesE + wpb - 1) / wpb, 4096);
  lin_edge_kernel<<<gridE, blk, 0, stream>>>(
      x, row, col, We1, be1, We2, be2, Wn11, bn11, Wn12, bn12, aggS, aggC, E,
      nTilesE);

  int nTilesN = (Nn + 15) / 16;
  int gridN = imin((nTilesN + wpb - 1) / wpb, 2048);
  lin_node_kernel<<<gridN, blk, 0, stream>>>(
      x, batch, aggS, aggC, Wn21, bn21, Wn22, bn22, gS, gC, Nn, nTilesN);

  int gridG = (G + 255) / 256;
  lin_global_kernel<<<gridG, blk, 0, stream>>>(gS, gC, Wg1, bg1, Wg2, bg2,
                                               (float*)d_out, G);
}